// VersorOdometry_74655121539761
// MI455X (gfx1250) — compile-verified
//
#include <hip/hip_runtime.h>

typedef __attribute__((ext_vector_type(2))) float v2f;
typedef __attribute__((ext_vector_type(8))) float v8f;

#define S_LEN 2048
#define TCH   128                  // timesteps per LDS chunk
#define NCHUNK (S_LEN / TCH)

// ---- Cl(4,1) sign table: bit i of v[k] set iff C[i, i^k, k] == -1 ----
struct SgnTab { unsigned v[32]; };
static constexpr SgnTab makeSgn() {
    SgnTab t{};
    for (int k = 0; k < 32; ++k) {
        unsigned w = 0u;
        for (int i = 0; i < 32; ++i) {
            const int a = i, b = i ^ k;
            int s = 0;
            int aa = a >> 1;
            while (aa) { s += __builtin_popcount(aa & b); aa >>= 1; }
            s += ((a & b) >> 4) & 1;        // metric: e4^2 = -1
            if (s & 1) w |= (1u << i);
        }
        t.v[k] = w;
    }
    return t;
}
constexpr SgnTab SGN_TAB = makeSgn();

// ---- VALU cross-lane reduction helpers (DPP16 + permlanex16, no LDS latency) ----
// quad_perm xor1 = 0xB1, quad_perm xor2 = 0x4E, row_ror:4 = 0x124, row_ror:8 = 0x128
#define DPP_ADD(p, ctrl)                                                        \
    (p) += __int_as_float(__builtin_amdgcn_update_dpp(                          \
        0, __float_as_int(p), (ctrl), 0xf, 0xf, true))

// 16-lane all-reduce sum (each half of the wave independently)
#define RED16(p) do {                                                           \
    DPP_ADD(p, 0xB1);  /* + xor1  */                                            \
    DPP_ADD(p, 0x4E);  /* + xor2  */                                            \
    DPP_ADD(p, 0x124); /* + ror4  */                                            \
    DPP_ADD(p, 0x128); /* + ror8  */                                            \
} while (0)

// cross-half exchange-add: lane i += lane i^16 (identity lane selects)
#define REDX16(p)                                                               \
    (p) += __int_as_float(__builtin_amdgcn_permlanex16(                         \
        0, __float_as_int(p), 0x76543210, 0xfedcba98, false, false))

__global__ __launch_bounds__(256)
void versor_odometry_kernel(const float* __restrict__ x,
                            const float* __restrict__ W_in,
                            const float* __restrict__ b_in,
                            const float* __restrict__ W_out,
                            const float* __restrict__ b_out,
                            float* __restrict__ out)
{
    __shared__ float psiBuf[TCH * 256];   // 128 KB: chunk of psi_seq (t-local, 256 cols)
    __shared__ float Wsw[256 * 32];       // 32 KB: W_out, pair-swizzled for b64 B-frag loads

    const int tid  = threadIdx.x;
    const int lane = tid & 31;
    const int h    = tid >> 5;            // wave id == head/channel id
    const int b    = blockIdx.x;
    const int lane4 = lane << 2;

    // Stage W_out into LDS: element (k,n) -> Wsw[(k>>1)*64 + 2n + (k&1)]
    for (int idx = tid; idx < 256 * 32; idx += 256) {
        const int k = idx >> 5, n = idx & 31;
        Wsw[(k >> 1) * 64 + (n << 1) + (k & 1)] = W_out[idx];
    }

    // Per-lane constants: lane holds blade component `lane` of chain (b,h)
    float Wc[6];
#pragma unroll
    for (int d = 0; d < 6; ++d) Wc[d] = W_in[d * 256 + (h << 5) + lane];
    const float bi   = b_in[(h << 5) + lane];
    const float bo0  = b_out[lane & 15];
    const float bo1  = b_out[(lane & 15) + 16];
    const float one0 = (lane == 0) ? 1.0f : 0.0f;
    const unsigned SG = SGN_TAB.v[lane];
    float sgnf[32];
#pragma unroll
    for (int i = 0; i < 32; ++i) sgnf[i] = ((SG >> i) & 1u) ? -1.0f : 1.0f;

    float psi = one0;                                  // identity rotor
    const float2* xrow = (const float2*)(x + (size_t)b * S_LEN * 6);

    __syncthreads();

    for (int c = 0; c < NCHUNK; ++c) {
        const int cbase = c * TCH;

        // ---------------- scan phase (serial critical path) ----------------
        float2 c0 = xrow[cbase * 3 + 0];
        float2 c1 = xrow[cbase * 3 + 1];
        float2 c2 = xrow[cbase * 3 + 2];
        for (int tm = 0; tm < TCH; ++tm) {
            const int t  = cbase + tm;
            const int tn = (t + 1 < S_LEN) ? (t + 1) : (S_LEN - 1);
            float2 n0 = xrow[tn * 3 + 0];              // prefetch next step's x
            float2 n1 = xrow[tn * 3 + 1];
            float2 n2 = xrow[tn * 3 + 2];

            // fused input projection: u_k = b_in_k + x . W_in[:,k]  (+1 on scalar blade)
            float u = bi;
            u = fmaf(c0.x, Wc[0], u); u = fmaf(c0.y, Wc[1], u);
            u = fmaf(c1.x, Wc[2], u); u = fmaf(c1.y, Wc[3], u);
            u = fmaf(c2.x, Wc[4], u); u = fmaf(c2.y, Wc[5], u);
            u += one0;
            // (delta normalization dropped: normalize(geo(.)) is scale-invariant)

            // geometric product: out_k = sum_i sgn(i,i^k) * u_i * psi_{i^k}
            const int psib = __float_as_int(psi);
            const int ub   = __float_as_int(u);
            float a0 = 0.f, a1 = 0.f, a2 = 0.f, a3 = 0.f;
#pragma unroll
            for (int i = 0; i < 32; ++i) {
                const float di = __int_as_float(__builtin_amdgcn_readlane(ub, i));
                const float pj = __int_as_float(
                    __builtin_amdgcn_ds_bpermute(lane4 ^ (i << 2), psib));
                const float term = sgnf[i] * pj;
                if      ((i & 3) == 0) a0 = fmaf(di, term, a0);
                else if ((i & 3) == 1) a1 = fmaf(di, term, a1);
                else if ((i & 3) == 2) a2 = fmaf(di, term, a2);
                else                   a3 = fmaf(di, term, a3);
            }
            const float v = (a0 + a1) + (a2 + a3);

            // wave-wide normalize of psi: 4 DPP adds + 1 permlanex16 (all VALU)
            float p = v * v;
            RED16(p);
            REDX16(p);
            const float inv = __builtin_amdgcn_rcpf(__builtin_amdgcn_sqrtf(p) + 1e-8f);
            psi = v * inv;

            psiBuf[tm * 256 + (h << 5) + lane] = psi;
            c0 = n0; c1 = n1; c2 = n2;
        }
        __syncthreads();

        // ------------- projection phase: (128 x 256) @ (256 x 32) via WMMA -------------
        // wave h owns 16-row tile [h*16, h*16+16); K-loop of V_WMMA_F32_16X16X4_F32
        const int row  = lane & 15;
        const int half = lane >> 4;                    // K sub-pair select per ISA A/B layout
        v8f acc0 = {0.f,0.f,0.f,0.f,0.f,0.f,0.f,0.f};
        v8f acc1 = {0.f,0.f,0.f,0.f,0.f,0.f,0.f,0.f};
        const float* aBase = psiBuf + (h * 16 + row) * 256 + half * 2;
        for (int kk = 0; kk < 64; ++kk) {
            const v2f A  = *(const v2f*)(aBase + kk * 4);
            const v2f B0 = *(const v2f*)(Wsw + (kk * 2 + half) * 64 + (row << 1));
            const v2f B1 = *(const v2f*)(Wsw + (kk * 2 + half) * 64 + (row << 1) + 32);
            acc0 = __builtin_amdgcn_wmma_f32_16x16x4_f32(
                false, A, false, B0, (short)0, acc0, false, false);
            acc1 = __builtin_amdgcn_wmma_f32_16x16x4_f32(
                false, A, false, B1, (short)0, acc1, false, false);
        }

        // bias, per-row normalize over N=32 (16-lane DPP reduce), store out[b,t,n]
#pragma unroll
        for (int j = 0; j < 8; ++j) {
            float y0 = acc0[j] + bo0;                  // N = row
            float y1 = acc1[j] + bo1;                  // N = row + 16
            float p  = fmaf(y0, y0, y1 * y1);
            RED16(p);                                  // halves are independent rows here
            const float inv = __builtin_amdgcn_rcpf(__builtin_amdgcn_sqrtf(p) + 1e-8f);
            y0 *= inv; y1 *= inv;
            const int t = cbase + h * 16 + j + (half << 3);
            float* o = out + ((((size_t)b << 11) + t) << 5) + row;
            o[0]  = y0;
            o[16] = y1;
        }
        __syncthreads();                               // psiBuf reused next chunk
    }
}

extern "C" void kernel_launch(void* const* d_in, const int* in_sizes, int n_in,
                              void* d_out, int out_size, void* d_ws, size_t ws_size,
                              hipStream_t stream) {
    (void)in_sizes; (void)n_in; (void)out_size; (void)d_ws; (void)ws_size;
    const float* x     = (const float*)d_in[0];
    const float* W_in  = (const float*)d_in[1];
    const float* b_in  = (const float*)d_in[2];
    const float* W_out = (const float*)d_in[3];
    const float* b_out = (const float*)d_in[4];
    float* out = (float*)d_out;
    versor_odometry_kernel<<<dim3(64), dim3(256), 0, stream>>>(
        x, W_in, b_in, W_out, b_out, out);
}